// ModifiedMarginRankingLoss_49744311222525
// MI455X (gfx1250) — compile-verified
//
#include <hip/hip_runtime.h>

// ModifiedMarginRankingLoss on gfx1250 (MI455X), sort-free rank formulation.
//   loss = (1/B) * sum_b sum_{pairs with r_b > r_a} relu(s_a - s_b + 0.1*(r_b - r_a))
// Identity: s_a - s_b + 0.1*(r_b - r_a) = q_a - q_b where q = s - 0.1*r,
// so the hinge body is one subtract + gated relu (fully branchless).
// Wave reductions use V_WMMA_F32_16X16X4_F32 (f32 in/out, no precision loss).

#define B_ROWS   32768
#define N_LEN    64
#define BLOCK    256
#define ROWS_PER_BLOCK (BLOCK / N_LEN)          // 4
#define NGROUPS  (B_ROWS / ROWS_PER_BLOCK)      // 8192 row-groups
#define NBLOCKS  2048                           // grid-stride over groups; 8 KB partials
#define MARGIN_F 0.1f

typedef __attribute__((ext_vector_type(2))) float v2f;
typedef __attribute__((ext_vector_type(8))) float v8f;

// Full wave32 sum via one f32 WMMA + ones-matrix trick.
// A layout (32-bit A 16x4): lane L<16 -> A[L][K=0] (a.x); lane L>=16 -> A[L-16][K=2].
// With B = ones(4x16): D[m][n] = rowsum[m] = x[m] + x[m+16], replicated over n.
// Lanes 0-15 hold rowsum[0..7] in d[0..7]; lanes 16-31 hold rowsum[8..15].
__device__ __forceinline__ float wave_sum_wmma(float x) {
    v2f a; a.x = x;    a.y = 0.0f;
    v2f b; b.x = 1.0f; b.y = 1.0f;   // every B slot = 1 regardless of layout
    v8f c = {};
    v8f d = __builtin_amdgcn_wmma_f32_16x16x4_f32(
        /*neg_a=*/false, a, /*neg_b=*/false, b,
        /*c_mod=*/(short)0, c, /*reuse_a=*/false, /*reuse_b=*/false);
    float t = d[0] + d[1] + d[2] + d[3] + d[4] + d[5] + d[6] + d[7];
    t += __shfl_xor(t, 16, 32);      // combine half-sums -> full wave sum in all lanes
    return t;
}

__global__ __launch_bounds__(BLOCK)
void rank_hinge_kernel(const float* __restrict__ scores,
                       const float* __restrict__ labels,
                       float* __restrict__ partial) {
    __shared__ float  sh_lab[BLOCK];
    __shared__ float2 sh_qr[BLOCK];         // (q = score - 0.1*rank, rank)
    __shared__ float  sh_wave[BLOCK / 32];

    const int tid   = threadIdx.x;
    const int lane  = tid & (N_LEN - 1);    // position within row
    const int rbase = tid & ~(N_LEN - 1);   // row base inside LDS

    float acc = 0.0f;
    for (int grp = blockIdx.x; grp < NGROUPS; grp += gridDim.x) {
        const int   g   = grp * BLOCK + tid;      // fully coalesced
        const float lab = labels[g];
        const float sc  = scores[g];

        __syncthreads();                          // prev iteration's LDS reads done
        sh_lab[tid] = lab;
        __syncthreads();

        // rank = #elements smaller (stable tie-break by original index)
        int r = 0;
        #pragma unroll 16
        for (int c2 = 0; c2 < N_LEN; ++c2) {
            const float lc = sh_lab[rbase + c2];  // wave-broadcast read: conflict-free
            r += (int)(lc < lab) + (int)((lc == lab) & (c2 < lane));
        }
        const float fr  = (float)r;
        const float myq = fmaf(-MARGIN_F, fr, sc);    // q_a = s_a - 0.1*r_a

        sh_qr[tid] = make_float2(myq, fr);
        __syncthreads();

        // hinge against all partners with larger rank (each pair counted once).
        // Branchless: select a negative sentinel when gate fails, relu kills it.
        #pragma unroll 16
        for (int c2 = 0; c2 < N_LEN; ++c2) {
            const float2 p  = sh_qr[rbase + c2];       // single ds_load_b64
            const float  d  = myq - p.x;               // q_a - q_b
            const float  dg = (p.y > fr) ? d : -1.0f;  // v_cndmask, no branch
            acc += fmaxf(dg, 0.0f);
        }
    }

    // block reduction: WMMA wave sums, then 8 values via LDS
    const float wsum = wave_sum_wmma(acc);
    if ((tid & 31) == 0) sh_wave[tid >> 5] = wsum;
    __syncthreads();
    if (tid == 0) {
        float s = 0.0f;
        #pragma unroll
        for (int i = 0; i < BLOCK / 32; ++i) s += sh_wave[i];
        partial[blockIdx.x] = s;
    }
}

__global__ __launch_bounds__(BLOCK)
void final_reduce_kernel(const float* __restrict__ partial,
                         float* __restrict__ out, int n, float scale) {
    __shared__ float sh_wave[BLOCK / 32];
    const int tid = threadIdx.x;
    float acc = 0.0f;
    for (int i = tid; i < n; i += BLOCK) acc += partial[i];   // uniform trip count
    const float wsum = wave_sum_wmma(acc);
    if ((tid & 31) == 0) sh_wave[tid >> 5] = wsum;
    __syncthreads();
    if (tid == 0) {
        float s = 0.0f;
        #pragma unroll
        for (int i = 0; i < BLOCK / 32; ++i) s += sh_wave[i];
        out[0] = s * scale;
    }
}

extern "C" void kernel_launch(void* const* d_in, const int* in_sizes, int n_in,
                              void* d_out, int out_size, void* d_ws, size_t ws_size,
                              hipStream_t stream) {
    const float* scores = (const float*)d_in[0];
    const float* labels = (const float*)d_in[1];
    float* out     = (float*)d_out;
    float* partial = (float*)d_ws;            // NBLOCKS floats (8 KB) of scratch

    rank_hinge_kernel<<<NBLOCKS, BLOCK, 0, stream>>>(scores, labels, partial);
    final_reduce_kernel<<<1, BLOCK, 0, stream>>>(partial, out, NBLOCKS,
                                                 1.0f / (float)B_ROWS);
}